// HeCoGATConv_1185410973814
// MI455X (gfx1250) — compile-verified
//
#include <hip/hip_runtime.h>
#include <hip/hip_bf16.h>
#include <math.h>

// GAT edge-softmax aggregation for MI455X (gfx1250, wave32).
//
// Memory-bound scatter/gather problem: dominant cost is E*D gather of
// feat_src (~819MB) + float atomics into out (51MB, L2-resident on the
// 192MB L2). el/er are real GEMVs -> computed with V_WMMA_F32_16X16X4_F32
// at full f32 precision.

#define NEG_SLOPE 0.01f

typedef __attribute__((ext_vector_type(2))) float v2f;
typedef __attribute__((ext_vector_type(8))) float v8f;

// Order-preserving float -> uint encoding (for atomicMax on signed floats).
// All finite/inf floats encode to >= 0x007FFFFF, so a 0-initialized cell acts
// as -inf; decoding 0 yields NaN which we guard like the reference does.
__device__ __forceinline__ unsigned encf(float f) {
  unsigned u = __float_as_uint(f);
  return (u & 0x80000000u) ? ~u : (u | 0x80000000u);
}
__device__ __forceinline__ float decf(unsigned u) {
  unsigned b = (u & 0x80000000u) ? (u ^ 0x80000000u) : ~u;
  return __uint_as_float(b);
}

__global__ void gat_init(float* __restrict__ out, unsigned* __restrict__ menc,
                         float* __restrict__ denom, int nOut, int nNd) {
  int stride = gridDim.x * blockDim.x;
  int tid = blockIdx.x * blockDim.x + threadIdx.x;
  for (int j = tid; j < nOut; j += stride) out[j] = 0.0f;
  for (int j = tid; j < nNd; j += stride) { menc[j] = 0u; denom[j] = 0.0f; }
}

// el[i] = sum_k feat[i,k]*attn[k], one 16-node tile per wave via WMMA f32.
// A-matrix 16x4 f32 layout (ISA 7.12.2): lanes 0-15 hold K=0 (vgpr0) / K=1
// (vgpr1); lanes 16-31 hold K=2 / K=3. B is attn replicated across all 16
// columns, so D[m, n] == el[m] for every n.
__global__ void gat_dots_wmma(const float* __restrict__ feat,
                              const float* __restrict__ attn,
                              float* __restrict__ outv, int n, int D) {
  int wave = (int)((blockIdx.x * (unsigned)blockDim.x + threadIdx.x) >> 5);
  int lane = (int)(threadIdx.x & 31u);
  int base = wave * 16;
  if (base >= n) return;            // wave-uniform: EXEC stays all-ones
  int half = lane >> 4;             // 0: K pair {0,1}; 1: K pair {2,3}
  int m16 = lane & 15;
  int row = base + m16; if (row > n - 1) row = n - 1;   // branchless clamp
  const float* rp = feat + (size_t)row * (size_t)D;

  v8f acc = {};
  for (int k0 = 0; k0 < D; k0 += 4) {
    v2f a, b;
    a.x = rp[k0 + 2 * half];
    a.y = rp[k0 + 2 * half + 1];
    b.x = attn[k0 + 2 * half];
    b.y = attn[k0 + 2 * half + 1];
    acc = __builtin_amdgcn_wmma_f32_16x16x4_f32(
        /*neg_a=*/false, a, /*neg_b=*/false, b,
        /*c_mod=*/(short)0, acc, /*reuse_a=*/false, /*reuse_b=*/false);
  }
  // D layout: VGPR v, lanes 0-15 -> M=v; lanes 16-31 -> M=v+8. Column n=0
  // lives in lane 0 (M=0..7) and lane 16 (M=8..15).
  if (m16 == 0) {
#pragma unroll
    for (int v = 0; v < 8; ++v) {
      int r = base + half * 8 + v;
      if (r < n) outv[r] = acc[v];
    }
  }
}

__global__ void gat_edge_logits(const int* __restrict__ src, const int* __restrict__ dst,
                                const float* __restrict__ el, const float* __restrict__ er,
                                float* __restrict__ e, unsigned* __restrict__ menc, int E) {
  int i = blockIdx.x * blockDim.x + threadIdx.x;
  if (i >= E) return;
  float v = el[src[i]] + er[dst[i]];
  v = (v >= 0.0f) ? v : NEG_SLOPE * v;   // leaky_relu
  e[i] = v;
  atomicMax(&menc[dst[i]], encf(v));     // segment max
}

__global__ void gat_edge_exp(const int* __restrict__ dst, const unsigned* __restrict__ menc,
                             float* __restrict__ e, float* __restrict__ denom, int E) {
  int i = blockIdx.x * blockDim.x + threadIdx.x;
  if (i >= E) return;
  int d = dst[i];
  float m = decf(menc[d]);
  if (!__builtin_isfinite(m)) m = 0.0f;  // reference's empty-segment guard
  float a = expf(e[i] - m);
  e[i] = a;                              // overwrite logit with unnormalized weight
  unsafeAtomicAdd(&denom[d], a);         // native global_atomic_add_f32
}

// One wave per edge; 32 lanes x float4 covers D=128 in one pass.
__global__ void gat_scatter(const int* __restrict__ src, const int* __restrict__ dst,
                            const float* __restrict__ a, const float* __restrict__ denom,
                            const float* __restrict__ feat_src, float* __restrict__ out,
                            int E, int D) {
  int wave = (int)((blockIdx.x * (unsigned)blockDim.x + threadIdx.x) >> 5);
  int lane = (int)(threadIdx.x & 31u);
  if (wave >= E) return;
  int s = src[wave];
  int d = dst[wave];
  float dn = denom[d];
  float w = a[wave] / (dn > 0.0f ? dn : 1.0f);
  const float* srow = feat_src + (size_t)s * (size_t)D;
  float* orow = out + (size_t)d * (size_t)D;
  for (int c = lane * 4; c < D; c += 128) {
    float4 f = *(const float4*)(srow + c);
    unsafeAtomicAdd(orow + c + 0, f.x * w);
    unsafeAtomicAdd(orow + c + 1, f.y * w);
    unsafeAtomicAdd(orow + c + 2, f.z * w);
    unsafeAtomicAdd(orow + c + 3, f.w * w);
  }
}

extern "C" void kernel_launch(void* const* d_in, const int* in_sizes, int n_in,
                              void* d_out, int out_size, void* d_ws, size_t ws_size,
                              hipStream_t stream) {
  const float* feat_src = (const float*)d_in[0];
  const float* feat_dst = (const float*)d_in[1];
  const int*   src      = (const int*)d_in[2];
  const int*   dst      = (const int*)d_in[3];
  const float* attn_l   = (const float*)d_in[4];
  const float* attn_r   = (const float*)d_in[5];

  const int D  = in_sizes[4];        // 128
  const int Ns = in_sizes[0] / D;    // 100000
  const int Nd = in_sizes[1] / D;    // 100000
  const int E  = in_sizes[2];        // 1600000
  float* out = (float*)d_out;

  // Workspace layout (~8 MB): el[Ns] | er[Nd] | e[E] | m_enc[Nd] | denom[Nd]
  float*    el    = (float*)d_ws;
  float*    er    = el + Ns;
  float*    ebuf  = er + Nd;
  unsigned* menc  = (unsigned*)(ebuf + E);
  float*    denom = (float*)(menc + Nd);

  gat_init<<<2048, 256, 0, stream>>>(out, menc, denom, Nd * D, Nd);

  const int dotsThreads = 256;                 // 8 waves -> 8 tiles -> 128 nodes/block
  const int nodesPerBlock = (dotsThreads / 32) * 16;
  gat_dots_wmma<<<(Ns + nodesPerBlock - 1) / nodesPerBlock, dotsThreads, 0, stream>>>(
      feat_src, attn_l, el, Ns, D);
  gat_dots_wmma<<<(Nd + nodesPerBlock - 1) / nodesPerBlock, dotsThreads, 0, stream>>>(
      feat_dst, attn_r, er, Nd, D);

  gat_edge_logits<<<(E + 255) / 256, 256, 0, stream>>>(src, dst, el, er, ebuf, menc, E);
  gat_edge_exp   <<<(E + 255) / 256, 256, 0, stream>>>(dst, menc, ebuf, denom, E);

  // one wave per edge: 8 edges per 256-thread block
  gat_scatter<<<(E + 7) / 8, 256, 0, stream>>>(src, dst, ebuf, denom, feat_src, out, E, D);
}